// GCN_41850161332531
// MI455X (gfx1250) — compile-verified
//
#include <hip/hip_runtime.h>
#include <hip/hip_bf16.h>

#define NNODES 100000
#define NEDGES 1000000
#define IN_C   128
#define HID_C  64
#define OUT_C  32

typedef __attribute__((ext_vector_type(2))) float v2f;
typedef __attribute__((ext_vector_type(8))) float v8f;

// ---------------------------------------------------------------------------
// 1) Per-call init: deg = 1.0 (self-loop contribution), zero agg1 and d_out.
//    (harness poisons ws/out with 0xAA and we accumulate with atomics, so
//    this must run at the start of every launch)
// ---------------------------------------------------------------------------
__global__ void gcn_init(float* __restrict__ deg,
                         float4* __restrict__ agg1,   // NNODES * HID_C floats
                         float4* __restrict__ out)    // NNODES * OUT_C floats
{
    int i = blockIdx.x * blockDim.x + threadIdx.x;
    if (i >= NNODES) return;
    deg[i] = 1.0f;
    float4 z = make_float4(0.f, 0.f, 0.f, 0.f);
#pragma unroll
    for (int q = 0; q < HID_C / 4; ++q) agg1[(size_t)i * (HID_C / 4) + q] = z;
#pragma unroll
    for (int q = 0; q < OUT_C / 4; ++q) out[(size_t)i * (OUT_C / 4) + q] = z;
}

// deg[dst] += 1 for each real edge (self-loops already counted in init)
__global__ void gcn_degree(const int* __restrict__ dst, float* __restrict__ deg)
{
    int e = blockIdx.x * blockDim.x + threadIdx.x;
    if (e >= NEDGES) return;
    atomicAdd(&deg[dst[e]], 1.0f);
}

// deg >= 1 always (self loops), so dinv = rsqrt(deg); matches reference's
// where(deg>0, rsqrt(max(deg,1)), 0).
__global__ void gcn_rsqrt(float* __restrict__ deg)
{
    int i = blockIdx.x * blockDim.x + threadIdx.x;
    if (i >= NNODES) return;
    deg[i] = rsqrtf(deg[i]);
}

// ---------------------------------------------------------------------------
// 2) fp32 WMMA GEMM: C[M,NC] = A[M,K] @ B[K,NC], one wave per 16x16 tile.
//    V_WMMA_F32_16X16X4_F32, K stepped by 4.
//    Fragment layouts per CDNA5 ISA 7.12.2 (32-bit matrices, wave32):
//      A 16x4 : lane(lo=lane&15,hi=lane>>4) vgpr g -> A[lo][2*hi+g]
//      B 4x16 : vgpr g, lanes lo -> B[2*hi+g][lo]
//      C 16x16: vgpr g -> C[g+8*hi][lo]
// ---------------------------------------------------------------------------
template <int K, int NC>
__global__ void gcn_gemm_wmma(const float* __restrict__ A,
                              const float* __restrict__ B,
                              float* __restrict__ C, int M)
{
    const int wave = (blockIdx.x * blockDim.x + threadIdx.x) >> 5;
    constexpr int TN = NC / 16;
    const int numTiles = (M >> 4) * TN;
    if (wave >= numTiles) return;          // wave-uniform guard; EXEC stays all-1s

    const int lane = threadIdx.x & 31;
    const int lo = lane & 15;
    const int hi = lane >> 4;
    const int tm = (wave / TN) << 4;       // tile row base
    const int tn = (wave % TN) << 4;       // tile col base

    const float* arow = A + (size_t)(tm + lo) * K + 2 * hi;   // A[m][2*hi + k...]
    const float* bcol = B + (size_t)(2 * hi) * NC + tn + lo;  // B[2*hi + k...][n]

    v8f acc = {};
#pragma unroll
    for (int k = 0; k < K; k += 4) {
        v2f a;
        a.x = arow[k];                     // A[m][k + 2*hi]     (merges to b64)
        a.y = arow[k + 1];                 // A[m][k + 2*hi + 1]
        v2f b;
        b.x = bcol[(size_t)k * NC];        // B[k + 2*hi][n]
        b.y = bcol[(size_t)(k + 1) * NC];  // B[k + 2*hi + 1][n]
        acc = __builtin_amdgcn_wmma_f32_16x16x4_f32(
            /*neg_a=*/false, a, /*neg_b=*/false, b,
            /*c_mod=*/(short)0, acc, /*reuse_a=*/false, /*reuse_b=*/false);
    }

    float* crow = C + (size_t)(tm + 8 * hi) * NC + tn + lo;
#pragma unroll
    for (int g = 0; g < 8; ++g) crow[(size_t)g * NC] = acc[g];
}

// ---------------------------------------------------------------------------
// 3) Edge scatter: out[dst] += h[src] * dinv[src]*dinv[dst].
//    One thread handles 4 channels of one edge (float4 gather + 4 atomics).
//    Edges e >= NEDGES are the self loops (src = dst = e - NEDGES).
// ---------------------------------------------------------------------------
template <int C>
__global__ void gcn_scatter(const float* __restrict__ h,
                            const int* __restrict__ src,
                            const int* __restrict__ dst,
                            const float* __restrict__ dinv,
                            float* __restrict__ out)
{
    constexpr int Q = C / 4;
    long long t = (long long)blockIdx.x * blockDim.x + threadIdx.x;
    if (t >= (long long)(NEDGES + NNODES) * Q) return;
    int e = (int)(t / Q);
    int q = (int)(t % Q);
    int s, d;
    if (e < NEDGES) { s = src[e]; d = dst[e]; }
    else            { s = e - NEDGES; d = s; }
    float w = dinv[s] * dinv[d];
    float4 hv = *(const float4*)(h + (size_t)s * C + 4 * q);
    float* o = out + (size_t)d * C + 4 * q;
    atomicAdd(o + 0, hv.x * w);
    atomicAdd(o + 1, hv.y * w);
    atomicAdd(o + 2, hv.z * w);
    atomicAdd(o + 3, hv.w * w);
}

// agg = relu(agg + b1), in place (input to layer-2 GEMM)
__global__ void gcn_bias_relu(float* __restrict__ agg, const float* __restrict__ b1)
{
    int i = blockIdx.x * blockDim.x + threadIdx.x;
    if (i >= NNODES * HID_C) return;
    float v = agg[i] + b1[i & (HID_C - 1)];
    agg[i] = v > 0.f ? v : 0.f;
}

// out += b2 (final bias)
__global__ void gcn_bias2(float* __restrict__ out, const float* __restrict__ b2)
{
    int i = blockIdx.x * blockDim.x + threadIdx.x;
    if (i >= NNODES * OUT_C) return;
    out[i] += b2[i & (OUT_C - 1)];
}

// ---------------------------------------------------------------------------
extern "C" void kernel_launch(void* const* d_in, const int* in_sizes, int n_in,
                              void* d_out, int out_size, void* d_ws, size_t ws_size,
                              hipStream_t stream)
{
    const float* x    = (const float*)d_in[0];               // [N,128]
    const int*   eidx = (const int*)d_in[1];                 // [2,E]
    const int*   src  = eidx;
    const int*   dst  = eidx + NEDGES;
    const float* W1   = (const float*)d_in[2];               // [128,64]
    const float* b1   = (const float*)d_in[3];               // [64]
    const float* W2   = (const float*)d_in[4];               // [64,32]
    const float* b2   = (const float*)d_in[5];               // [32]
    float* out = (float*)d_out;                              // [N,32]

    // workspace layout
    float* dinv = (float*)d_ws;                              // N floats (deg->dinv)
    float* h    = dinv + NNODES;                             // N*64 floats (h1, then h2)
    float* agg1 = h + (size_t)NNODES * HID_C;                // N*64 floats

    const int B = 256;

    // 1) init deg=1, zero agg1 and out
    gcn_init<<<(NNODES + B - 1) / B, B, 0, stream>>>(dinv, (float4*)agg1, (float4*)out);

    // 2) degrees and dinv
    gcn_degree<<<(NEDGES + B - 1) / B, B, 0, stream>>>(dst, dinv);
    gcn_rsqrt<<<(NNODES + B - 1) / B, B, 0, stream>>>(dinv);

    // 3) h = x @ W1   (fp32 WMMA)
    {
        int waves = (NNODES / 16) * (HID_C / 16);            // 25000
        int thr = waves * 32;
        gcn_gemm_wmma<IN_C, HID_C><<<(thr + B - 1) / B, B, 0, stream>>>(x, W1, h, NNODES);
    }

    // 4) agg1 = scatter-add of normalized messages
    {
        long long thr = (long long)(NEDGES + NNODES) * (HID_C / 4);
        gcn_scatter<HID_C><<<(int)((thr + B - 1) / B), B, 0, stream>>>(h, src, dst, dinv, agg1);
    }

    // 5) agg1 = relu(agg1 + b1)
    gcn_bias_relu<<<(NNODES * HID_C + B - 1) / B, B, 0, stream>>>(agg1, b1);

    // 6) h2 = agg1 @ W2  (reuse h buffer, N*32 fits)
    {
        int waves = (NNODES / 16) * (OUT_C / 16);            // 12500
        int thr = waves * 32;
        gcn_gemm_wmma<HID_C, OUT_C><<<(thr + B - 1) / B, B, 0, stream>>>(agg1, W2, h, NNODES);
    }

    // 7) out = scatter-add of normalized layer-2 messages
    {
        long long thr = (long long)(NEDGES + NNODES) * (OUT_C / 4);
        gcn_scatter<OUT_C><<<(int)((thr + B - 1) / B), B, 0, stream>>>(h, src, dst, dinv, out);
    }

    // 8) out += b2
    gcn_bias2<<<(NNODES * OUT_C + B - 1) / B, B, 0, stream>>>(out, b2);
}